// FastWaveletTransform_5841155522677
// MI455X (gfx1250) — compile-verified
//
#include <hip/hip_runtime.h>

// ---------------------------------------------------------------------------
// 4-level 2D Haar DWT (pywt 'db1' analysis) for (64,3,512,512) f32.
//
// Per 16x16 tile X:   Y = W2 * X * W1^T
//   W1 (16x16, entries {0,+1,-1}): rows 0-7 sum column pairs, rows 8-15 diff.
//   W2 (16x16, entries {0,+-0.5}): rows 0-7 avg row pairs,    rows 8-15 diff/2.
//   Y quadrants: [cA cV; cH cD] with exactly the reference's *0.5 scaling.
// Computed on the WMMA pipe as two chained f32 16x16x16 matmuls:
//   P = X * W1^T          (4 x v_wmma_f32_16x16x4_f32)
//   Z = P^T * W2^T = Y^T  (4 x v_wmma_f32_16x16x4_f32)
// P^T is built from P's C/D layout with 4 ds_swizzle (SWAPX16) + cndmasks.
// Memory-bound workload: 1KB in / 1KB out per tile via b64 loads, b128 stores.
// ---------------------------------------------------------------------------

typedef __attribute__((ext_vector_type(2))) float v2f;
typedef __attribute__((ext_vector_type(8))) float v8f;

// W1^T[k][n] (and W2^T = 0.5 * W1^T): Haar pair-combining matrix.
__device__ __forceinline__ float haarColMat(int k, int n) {
  const int m = n & 7;                       // pair index
  float v = ((k >> 1) == m) ? 1.0f : 0.0f;   // nonzero at k = 2m, 2m+1
  if ((n >= 8) && (k & 1)) v = -v;           // high-pass rows: -1 on odd k
  return v;
}

#define SWZ_SWAPX16(x) \
  __int_as_float(__builtin_amdgcn_ds_swizzle(__float_as_int(x), 0x401F))

__global__ __launch_bounds__(256) void haar_dwt2_wmma(
    const float* __restrict__ src, unsigned long long srcPlaneStride,
    float* __restrict__ dst, int W, int H, int nPlanes)
{
  const int lane = threadIdx.x & 31;
  const int n    = lane & 15;   // matrix column index (B/N, C/D lane dim)
  const int hi   = lane >> 4;   // which lane-half (K split)
  const int nTX  = W >> 4;
  const int nTY  = H >> 4;
  const int tilesPerPlane = nTX * nTY;

  const int wave = blockIdx.x * (blockDim.x >> 5) + (threadIdx.x >> 5);
  if (wave >= tilesPerPlane * nPlanes) return;   // wave-uniform: EXEC stays all-1s

  const int plane = wave / tilesPerPlane;
  const int trem  = wave - plane * tilesPerPlane;
  const int ty    = trem / nTX;
  const int tx    = trem - ty * nTX;

  // A-operand layout (f32, 16x4 per step): lane holds X[row=n, K=4kk+2*hi+{0,1}]
  // -> contiguous float2 loads.
  const float* srow = src + (unsigned long long)plane * srcPlaneStride
                    + (unsigned long long)(ty * 16 + n) * (unsigned)W
                    + (unsigned)(tx * 16 + 2 * hi);

  const float2 x0 = *(const float2*)(srow + 0);
  const float2 x1 = *(const float2*)(srow + 4);
  const float2 x2 = *(const float2*)(srow + 8);
  const float2 x3 = *(const float2*)(srow + 12);

  __builtin_prefetch(srow + 16, 0, 1);  // warm next tile (global_prefetch_b8)

  const int kb = 2 * hi;

  // -------- pass 1: P = X * W1^T (combine column pairs) --------
  v8f p = {0.f, 0.f, 0.f, 0.f, 0.f, 0.f, 0.f, 0.f};
  {
    v2f a, b;
    a[0] = x0.x; a[1] = x0.y;
    b[0] = haarColMat(kb + 0, n);  b[1] = haarColMat(kb + 1, n);
    p = __builtin_amdgcn_wmma_f32_16x16x4_f32(false, a, false, b, (short)0, p, false, false);
    a[0] = x1.x; a[1] = x1.y;
    b[0] = haarColMat(kb + 4, n);  b[1] = haarColMat(kb + 5, n);
    p = __builtin_amdgcn_wmma_f32_16x16x4_f32(false, a, false, b, (short)0, p, false, false);
    a[0] = x2.x; a[1] = x2.y;
    b[0] = haarColMat(kb + 8, n);  b[1] = haarColMat(kb + 9, n);
    p = __builtin_amdgcn_wmma_f32_16x16x4_f32(false, a, false, b, (short)0, p, false, false);
    a[0] = x3.x; a[1] = x3.y;
    b[0] = haarColMat(kb + 12, n); b[1] = haarColMat(kb + 13, n);
    p = __builtin_amdgcn_wmma_f32_16x16x4_f32(false, a, false, b, (short)0, p, false, false);
  }

  // -------- half-wave exchange: build P^T's A-operand layout --------
  // Lane half holds P rows {8*hi .. 8*hi+7} of column n; it needs rows
  // {4kk+2*hi, 4kk+2*hi+1}. Rows 8,9,12,13 (lo) / 2,3,6,7 (hi) come from the
  // partner lane via SWAPX16 swizzle.
  const bool h = (hi != 0);
  const float t0 = h ? p[0] : p[2];
  const float t1 = h ? p[1] : p[3];
  const float t2 = h ? p[4] : p[6];
  const float t3 = h ? p[5] : p[7];
  const float s0 = SWZ_SWAPX16(t0);
  const float s1 = SWZ_SWAPX16(t1);
  const float s2 = SWZ_SWAPX16(t2);
  const float s3 = SWZ_SWAPX16(t3);

  // -------- pass 2: Z = P^T * W2^T  ( = (W2 * P)^T = Y^T ) --------
  v8f z = {0.f, 0.f, 0.f, 0.f, 0.f, 0.f, 0.f, 0.f};
  {
    v2f a, b;
    a[0] = h ? s0 : p[0];  a[1] = h ? s1 : p[1];            // rows 2hi, 2hi+1
    b[0] = 0.5f * haarColMat(kb + 0, n);  b[1] = 0.5f * haarColMat(kb + 1, n);
    z = __builtin_amdgcn_wmma_f32_16x16x4_f32(false, a, false, b, (short)0, z, false, false);
    a[0] = h ? s2 : p[4];  a[1] = h ? s3 : p[5];            // rows 4+2hi, 5+2hi
    b[0] = 0.5f * haarColMat(kb + 4, n);  b[1] = 0.5f * haarColMat(kb + 5, n);
    z = __builtin_amdgcn_wmma_f32_16x16x4_f32(false, a, false, b, (short)0, z, false, false);
    a[0] = h ? p[2] : s0;  a[1] = h ? p[3] : s1;            // rows 8+2hi, 9+2hi
    b[0] = 0.5f * haarColMat(kb + 8, n);  b[1] = 0.5f * haarColMat(kb + 9, n);
    z = __builtin_amdgcn_wmma_f32_16x16x4_f32(false, a, false, b, (short)0, z, false, false);
    a[0] = h ? p[6] : s2;  a[1] = h ? p[7] : s3;            // rows 12+2hi, 13+2hi
    b[0] = 0.5f * haarColMat(kb + 12, n); b[1] = 0.5f * haarColMat(kb + 13, n);
    z = __builtin_amdgcn_wmma_f32_16x16x4_f32(false, a, false, b, (short)0, z, false, false);
  }

  // -------- store: lane holds Y[n, 8*hi .. 8*hi+7] = 8 consecutive floats of
  // one subband row.  Subband: cA(n<8,hi=0) cH(n>=8,hi=0) cV(n<8,hi=1) cD(else)
  const int halfW = W >> 1;
  const int halfH = H >> 1;
  const int sb = ((n >> 3) & 1) | (hi << 1);
  float* dp = dst
            + ((unsigned long long)plane * 4ull + (unsigned)sb)
              * (unsigned long long)(halfW * halfH)
            + (unsigned long long)(ty * 8 + (n & 7)) * (unsigned)halfW
            + (unsigned)(tx * 8);
  *(float4*)(dp + 0) = make_float4(z[0], z[1], z[2], z[3]);
  *(float4*)(dp + 4) = make_float4(z[4], z[5], z[6], z[7]);
}

extern "C" void kernel_launch(void* const* d_in, const int* in_sizes, int n_in,
                              void* d_out, int out_size, void* d_ws, size_t ws_size,
                              hipStream_t stream) {
  const float* x = (const float*)d_in[0];
  float* out = (float*)d_out;

  int planes = (n_in > 0) ? (in_sizes[0] / (512 * 512)) : 0;
  if (planes <= 0) planes = 64 * 3;

  // Output layout: levels concatenated; level L is [planes][4][H/2][W/2].
  const unsigned long long o1 = 0ull;
  const unsigned long long o2 = o1 + (unsigned long long)planes * 4ull * 256 * 256;
  const unsigned long long o3 = o2 + (unsigned long long)planes * 4ull * 128 * 128;
  const unsigned long long o4 = o3 + (unsigned long long)planes * 4ull * 64 * 64;

  struct Lvl { const float* s; unsigned long long sps; float* d; int W; };
  const Lvl lv[4] = {
    { x,        512ull * 512,      out + o1, 512 },
    { out + o1, 4ull * 256 * 256,  out + o2, 256 },   // cA is subband 0 of prev level
    { out + o2, 4ull * 128 * 128,  out + o3, 128 },
    { out + o3, 4ull * 64 * 64,    out + o4, 64  },
  };

  for (int i = 0; i < 4; ++i) {
    const int W = lv[i].W;
    const int waves  = (W / 16) * (W / 16) * planes;  // one wave32 per 16x16 tile
    const int blocks = (waves + 7) / 8;               // 8 waves per 256-thread block
    haar_dwt2_wmma<<<blocks, 256, 0, stream>>>(lv[i].s, lv[i].sps, lv[i].d,
                                               W, W, planes);
  }
}